// EdgeGATv2Conv_69312182223711
// MI455X (gfx1250) — compile-verified
//
#include <hip/hip_runtime.h>
#include <hip/hip_bf16.h>
#include <math.h>

#define N_NODES 100000
#define N_EDGES 1600000
#define IN_FEATS 128
#define EDGE_FEATS 64
#define NUM_HEADS 4
#define OUT_FEATS 16
#define HF 64
#define NEG_SLOPE 0.2f

#define NT_EDGE (N_EDGES / 16)
#define BLK_E 128                 // 4 wave32 per block for edge kernels
#define WPB_E (BLK_E / 32)
#define TILE_BYTES 4096           // 16 edges x 64 f32
#define EDGE_BLOCKS 2048          // persistent grid: 8192 waves, ~12 tiles/wave

typedef float v2f __attribute__((ext_vector_type(2)));
typedef float v8f __attribute__((ext_vector_type(8)));

// sign-aware float atomic max (init must be -inf)
__device__ __forceinline__ void atomicMaxFloat(float* addr, float val) {
    if (val >= 0.0f) {
        atomicMax((int*)addr, __float_as_int(val));
    } else {
        atomicMin((unsigned int*)addr, __float_as_uint(val));
    }
}

// low 32 bits of a flat pointer into LDS == LDS byte offset (aperture is addr[63:32])
__device__ __forceinline__ unsigned lds_addr32(const void* p) {
    return (unsigned)(unsigned long long)p;
}

// Copy one contiguous 4096B tile global->LDS with 8 async B128 ops.
// Each op: 32 lanes x 16B = 512B, fully coalesced; the same IOFFSET is applied
// to both the LDS destination and the global source, so one lane-base pair
// plus immediate offsets walks the whole tile. Tracked on ASYNCcnt.
__device__ __forceinline__ void async_tile_load(unsigned loff, const void* gsrc) {
    asm volatile(
        "global_load_async_to_lds_b128 %0, %1, off\n\t"
        "global_load_async_to_lds_b128 %0, %1, off offset:512\n\t"
        "global_load_async_to_lds_b128 %0, %1, off offset:1024\n\t"
        "global_load_async_to_lds_b128 %0, %1, off offset:1536\n\t"
        "global_load_async_to_lds_b128 %0, %1, off offset:2048\n\t"
        "global_load_async_to_lds_b128 %0, %1, off offset:2560\n\t"
        "global_load_async_to_lds_b128 %0, %1, off offset:3072\n\t"
        "global_load_async_to_lds_b128 %0, %1, off offset:3584"
        :: "v"(loff), "v"(gsrc) : "memory");
}

__global__ void init_kernel(float* mx, float* denom, float* out) {
    int i = blockIdx.x * blockDim.x + threadIdx.x;
    if (i < N_NODES * NUM_HEADS) {
        mx[i] = -INFINITY;
        denom[i] = 0.0f;
    }
    if (i < N_NODES * HF) out[i] = 0.0f;
}

// el = feat @ W_src + b_src ; er = feat @ W_dst + b_dst
// one wave per 16-row tile; all 4 col-tiles for both weight matrices.
__global__ void node_proj_kernel(const float* __restrict__ feat,
                                 const float* __restrict__ Wsrc, const float* __restrict__ bsrc,
                                 const float* __restrict__ Wdst, const float* __restrict__ bdst,
                                 float* __restrict__ el, float* __restrict__ er) {
    int wave = (blockIdx.x * blockDim.x + threadIdx.x) >> 5;
    int lane = threadIdx.x & 31;
    if (wave >= N_NODES / 16) return;          // wave-uniform: EXEC all-1s inside
    int r0 = wave * 16;
    int n = lane & 15;                         // A-row / D-col index
    int h = lane >> 4;                         // half-wave

    v8f accS[4] = {};
    v8f accD[4] = {};

    for (int k0 = 0; k0 < IN_FEATS; k0 += 4) {
        const float* ap = feat + (size_t)(r0 + n) * IN_FEATS + k0 + 2 * h;
        v2f a; a.x = ap[0]; a.y = ap[1];
#pragma unroll
        for (int ct = 0; ct < 4; ++ct) {
            int c0 = ct * 16;
            const float* bpS = Wsrc + (size_t)(k0 + 2 * h) * HF + c0 + n;
            const float* bpD = Wdst + (size_t)(k0 + 2 * h) * HF + c0 + n;
            v2f bS; bS.x = bpS[0]; bS.y = bpS[HF];
            v2f bD; bD.x = bpD[0]; bD.y = bpD[HF];
            accS[ct] = __builtin_amdgcn_wmma_f32_16x16x4_f32(false, a, false, bS,
                                                             (short)0, accS[ct], false, false);
            accD[ct] = __builtin_amdgcn_wmma_f32_16x16x4_f32(false, a, false, bD,
                                                             (short)0, accD[ct], false, false);
        }
    }
#pragma unroll
    for (int ct = 0; ct < 4; ++ct) {
        int c0 = ct * 16;
        float biasS = bsrc[c0 + n];
        float biasD = bdst[c0 + n];
#pragma unroll
        for (int v = 0; v < 8; ++v) {
            int row = r0 + v + 8 * h;          // D-row = vgpr + 8*half
            el[(size_t)row * HF + c0 + n] = accS[ct][v] + biasS;
            er[(size_t)row * HF + c0 + n] = accD[ct][v] + biasD;
        }
    }
}

// ee = edge_feat @ W_edge + b ; e = lrelu(el[src]+er[dst]+ee) ; score = <e, attn>
// persistent waves, LDS double-buffered async tile streaming.
__global__ void __launch_bounds__(BLK_E)
edge_score_kernel(const float* __restrict__ efeat,
                  const float* __restrict__ We, const float* __restrict__ be,
                  const int* __restrict__ src, const int* __restrict__ dst,
                  const float* __restrict__ el, const float* __restrict__ er,
                  const float* __restrict__ attn,
                  float* __restrict__ score, float* __restrict__ mx) {
    __shared__ __align__(16) float sW[EDGE_FEATS * HF];                   // 16 KB
    __shared__ __align__(16) char stile[WPB_E * 2 * TILE_BYTES];          // 32 KB

    for (int i = threadIdx.x; i < EDGE_FEATS * HF; i += BLK_E) sW[i] = We[i];
    __syncthreads();

    int wid = threadIdx.x >> 5;
    int lane = threadIdx.x & 31;
    int n = lane & 15;
    int h = lane >> 4;
    int gwave = blockIdx.x * WPB_E + wid;
    int nw = gridDim.x * WPB_E;

    float bias[4], attw[4];
#pragma unroll
    for (int ct = 0; ct < 4; ++ct) {
        bias[ct] = be[ct * 16 + n];
        attw[ct] = attn[ct * 16 + n];
    }

    char* mybuf = stile + wid * (2 * TILE_BYTES);
    unsigned lbase = lds_addr32(mybuf) + lane * 16;

    int t = gwave;
    if (t < NT_EDGE)
        async_tile_load(lbase, (const char*)(efeat + (size_t)t * 1024) + lane * 16);

    int buf = 0;
    for (; t < NT_EDGE; t += nw) {
        int tn = t + nw;
        if (tn < NT_EDGE) {   // wave-uniform branch
            async_tile_load(lbase + (buf ^ 1) * TILE_BYTES,
                            (const char*)(efeat + (size_t)tn * 1024) + lane * 16);
            asm volatile("s_wait_asynccnt 0x8" ::: "memory");  // current tile resident
        } else {
            asm volatile("s_wait_asynccnt 0x0" ::: "memory");
        }
        const float* tile = (const float*)(mybuf + buf * TILE_BYTES);

        v8f acc[4] = {};
#pragma unroll
        for (int kk = 0; kk < 16; ++kk) {
            v2f a;
            a.x = tile[n * EDGE_FEATS + kk * 4 + 2 * h];
            a.y = tile[n * EDGE_FEATS + kk * 4 + 2 * h + 1];
#pragma unroll
            for (int ct = 0; ct < 4; ++ct) {
                v2f b;
                b.x = sW[(kk * 4 + 2 * h) * HF + ct * 16 + n];
                b.y = sW[(kk * 4 + 2 * h + 1) * HF + ct * 16 + n];
                acc[ct] = __builtin_amdgcn_wmma_f32_16x16x4_f32(false, a, false, b,
                                                                (short)0, acc[ct], false, false);
            }
        }

        int e0 = t * 16;
#pragma unroll
        for (int v = 0; v < 8; ++v) {
            int eidx = e0 + v + 8 * h;
            int s = src[eidx];
            int d = dst[eidx];
#pragma unroll
            for (int ct = 0; ct < 4; ++ct) {
                float ev = acc[ct][v] + bias[ct]
                         + el[(size_t)s * HF + ct * 16 + n]
                         + er[(size_t)d * HF + ct * 16 + n];
                ev = (ev >= 0.0f) ? ev : NEG_SLOPE * ev;
                float p = ev * attw[ct];
                p += __shfl_xor(p, 1, 32);
                p += __shfl_xor(p, 2, 32);
                p += __shfl_xor(p, 4, 32);
                p += __shfl_xor(p, 8, 32);
                if (n == 0) {
                    score[(size_t)eidx * NUM_HEADS + ct] = p;
                    atomicMaxFloat(&mx[(size_t)d * NUM_HEADS + ct], p);
                }
            }
        }
        buf ^= 1;
    }
}

// ex = exp(score - mx[dst]) written in place over score; denom[dst] += ex
__global__ void exp_sum_kernel(const int* __restrict__ dst,
                               const float* __restrict__ mx,
                               float* __restrict__ score_ex,
                               float* __restrict__ denom) {
    long long i = (long long)blockIdx.x * blockDim.x + threadIdx.x;
    if (i >= (long long)N_EDGES * NUM_HEADS) return;
    int e = (int)(i >> 2);
    int hd = (int)(i & 3);
    int d = dst[e];
    float m = mx[(size_t)d * NUM_HEADS + hd];
    if (m == -INFINITY) m = 0.0f;              // matches reference's isfinite guard
    float ex = expf(score_ex[i] - m);
    score_ex[i] = ex;
    atomicAdd(&denom[(size_t)d * NUM_HEADS + hd], ex);
}

// recompute ee via WMMA; ft = el[src] + ee; out[dst] += ft * (ex / denom[dst])
__global__ void __launch_bounds__(BLK_E)
aggregate_kernel(const float* __restrict__ efeat,
                 const float* __restrict__ We, const float* __restrict__ be,
                 const int* __restrict__ src, const int* __restrict__ dst,
                 const float* __restrict__ el,
                 const float* __restrict__ ex, const float* __restrict__ denom,
                 float* __restrict__ out) {
    __shared__ __align__(16) float sW[EDGE_FEATS * HF];
    __shared__ __align__(16) char stile[WPB_E * 2 * TILE_BYTES];

    for (int i = threadIdx.x; i < EDGE_FEATS * HF; i += BLK_E) sW[i] = We[i];
    __syncthreads();

    int wid = threadIdx.x >> 5;
    int lane = threadIdx.x & 31;
    int n = lane & 15;
    int h = lane >> 4;
    int gwave = blockIdx.x * WPB_E + wid;
    int nw = gridDim.x * WPB_E;

    float bias[4];
#pragma unroll
    for (int ct = 0; ct < 4; ++ct) bias[ct] = be[ct * 16 + n];

    char* mybuf = stile + wid * (2 * TILE_BYTES);
    unsigned lbase = lds_addr32(mybuf) + lane * 16;

    int t = gwave;
    if (t < NT_EDGE)
        async_tile_load(lbase, (const char*)(efeat + (size_t)t * 1024) + lane * 16);

    int buf = 0;
    for (; t < NT_EDGE; t += nw) {
        int tn = t + nw;
        if (tn < NT_EDGE) {
            async_tile_load(lbase + (buf ^ 1) * TILE_BYTES,
                            (const char*)(efeat + (size_t)tn * 1024) + lane * 16);
            asm volatile("s_wait_asynccnt 0x8" ::: "memory");
        } else {
            asm volatile("s_wait_asynccnt 0x0" ::: "memory");
        }
        const float* tile = (const float*)(mybuf + buf * TILE_BYTES);

        v8f acc[4] = {};
#pragma unroll
        for (int kk = 0; kk < 16; ++kk) {
            v2f a;
            a.x = tile[n * EDGE_FEATS + kk * 4 + 2 * h];
            a.y = tile[n * EDGE_FEATS + kk * 4 + 2 * h + 1];
#pragma unroll
            for (int ct = 0; ct < 4; ++ct) {
                v2f b;
                b.x = sW[(kk * 4 + 2 * h) * HF + ct * 16 + n];
                b.y = sW[(kk * 4 + 2 * h + 1) * HF + ct * 16 + n];
                acc[ct] = __builtin_amdgcn_wmma_f32_16x16x4_f32(false, a, false, b,
                                                                (short)0, acc[ct], false, false);
            }
        }

        int e0 = t * 16;
#pragma unroll
        for (int v = 0; v < 8; ++v) {
            int eidx = e0 + v + 8 * h;
            int s = src[eidx];
            int d = dst[eidx];
#pragma unroll
            for (int ct = 0; ct < 4; ++ct) {
                float a = ex[(size_t)eidx * NUM_HEADS + ct]
                        / denom[(size_t)d * NUM_HEADS + ct];
                float ft = acc[ct][v] + bias[ct] + el[(size_t)s * HF + ct * 16 + n];
                atomicAdd(&out[(size_t)d * HF + ct * 16 + n], ft * a);
            }
        }
        buf ^= 1;
    }
}

extern "C" void kernel_launch(void* const* d_in, const int* in_sizes, int n_in,
                              void* d_out, int out_size, void* d_ws, size_t ws_size,
                              hipStream_t stream) {
    const float* feat  = (const float*)d_in[0];
    const float* efeat = (const float*)d_in[1];
    const int*   src   = (const int*)d_in[2];
    const int*   dst   = (const int*)d_in[3];
    const float* Wsrc  = (const float*)d_in[4];
    const float* bsrc  = (const float*)d_in[5];
    const float* Wdst  = (const float*)d_in[6];
    const float* bdst  = (const float*)d_in[7];
    const float* Wedge = (const float*)d_in[8];
    const float* bedge = (const float*)d_in[9];
    const float* attn  = (const float*)d_in[10];
    float* out = (float*)d_out;

    // workspace layout (~80 MB)
    float* el    = (float*)d_ws;                       // N*64
    float* er    = el + (size_t)N_NODES * HF;          // N*64
    float* scex  = er + (size_t)N_NODES * HF;          // E*4 (score, then ex in place)
    float* mx    = scex + (size_t)N_EDGES * NUM_HEADS; // N*4
    float* denom = mx + (size_t)N_NODES * NUM_HEADS;   // N*4

    const int BLK = 256;
    const int WPB = BLK / 32;

    {   // init mx/denom/out
        int total = N_NODES * HF;
        init_kernel<<<(total + BLK - 1) / BLK, BLK, 0, stream>>>(mx, denom, out);
    }
    {   // node projections
        int waves = N_NODES / 16;        // 6250
        node_proj_kernel<<<(waves + WPB - 1) / WPB, BLK, 0, stream>>>(
            feat, Wsrc, bsrc, Wdst, bdst, el, er);
    }
    {   // edge scores + segment max (persistent, async double-buffered)
        edge_score_kernel<<<EDGE_BLOCKS, BLK_E, 0, stream>>>(
            efeat, Wedge, bedge, src, dst, el, er, attn, scex, mx);
    }
    {   // exp + denom
        long long total = (long long)N_EDGES * NUM_HEADS;
        exp_sum_kernel<<<(int)((total + BLK - 1) / BLK), BLK, 0, stream>>>(
            dst, mx, scex, denom);
    }
    {   // weighted aggregation (persistent, async double-buffered)
        aggregate_kernel<<<EDGE_BLOCKS, BLK_E, 0, stream>>>(
            efeat, Wedge, bedge, src, dst, el, scex, denom, out);
    }
}